// Attention_25915832664752
// MI455X (gfx1250) — compile-verified
//
#include <hip/hip_runtime.h>
#include <hip/hip_bf16.h>
#include <cstddef>
#include <cstdint>

typedef __attribute__((ext_vector_type(2))) float v2f;
typedef __attribute__((ext_vector_type(8))) float v8f;
typedef __attribute__((ext_vector_type(4))) int   v4i;

// address-space-qualified v4i for the async global->LDS builtin
typedef __attribute__((address_space(1))) v4i ga_v4i;   // global
typedef __attribute__((address_space(3))) v4i la_v4i;   // LDS

#define CH   2048   // channels (nhid == nopen)
#define NSP  48     // spatial length N
#define KT   9      // conv taps
#define KTOT (CH * KT)      // 18432 = GEMM reduction length for the convs
#define NCH4 (KTOT / 4)     // 4608 K-chunks of 4
#define EPS  1e-5f

#if defined(__has_builtin)
#if __has_builtin(__builtin_amdgcn_global_load_async_to_lds_b128) && \
    __has_builtin(__builtin_amdgcn_s_wait_asynccnt)
#define HAVE_ASYNC_LDS 1
#endif
#endif

static __device__ __forceinline__ v8f wmma_f32x4(v2f a, v2f b, v8f c) {
  // V_WMMA_F32_16X16X4_F32 : D(16x16 f32) = A(16x4 f32) * B(4x16 f32) + C
  return __builtin_amdgcn_wmma_f32_16x16x4_f32(false, a, false, b, (short)0, c,
                                               false, false);
}

// ---------------------------------------------------------------------------
// Kernel 0: build im2col matrix in WMMA-B-fragment order.
//   Zcol logical: [k=0..18431][n=0..47],  Zcol[k][n] = Zpad[k/9][n + k%9 - 4]
//   Physical: fragment f = k4*3 + nt holds the 4x16 B tile for K-chunk k4,
//   n-tile nt, stored as 32 lanes x float2 (256 B, coalesced):
//     lane l, slot s  ->  B[2*(l>>4)+s][nt*16 + (l&15)]
// ---------------------------------------------------------------------------
__global__ __launch_bounds__(256) void im2col_kernel(
    const float* __restrict__ Z, float* __restrict__ Zcol) {
  const int gid = blockIdx.x * 256 + threadIdx.x;     // one per (frag, lane)
  const int total = NCH4 * 3 * 32;                    // 442368
  if (gid >= total) return;
  const int lanei = gid & 31;
  const int f     = gid >> 5;
  const int nt    = f % 3;
  const int k4    = f / 3;
  const int lh    = lanei >> 4;
  const int lm    = lanei & 15;
  const int col   = nt * 16 + lm;

  float2 v;
#pragma unroll
  for (int s = 0; s < 2; ++s) {
    const int k  = k4 * 4 + 2 * lh + s;
    const int ci = k / KT;
    const int t  = k - ci * KT;
    const int m  = col + t - 4;                       // same-padding shift
    float val = 0.f;
    if (m >= 0 && m < NSP) val = Z[ci * NSP + m];
    (s ? v.y : v.x) = val;
  }
  ((float2*)Zcol)[gid] = v;
}

// ---------------------------------------------------------------------------
// Kernel 1: six conv branches as one long-K WMMA GEMM each.
//   R[b][co][n] = sum_k W[b][co][k] * Zcol[k][n]   (K = 18432)
// grid = (CH/16, 6), block = 256 (8 waves, contiguous K split), LDS reduce.
// Inner loop: 1 A b64 + 3 B b64 + 3 WMMA per K-chunk of 4.
// ---------------------------------------------------------------------------
__global__ __launch_bounds__(256) void conv_wmma_kernel(
    const float* __restrict__ W, const float* __restrict__ Zcol,
    float* __restrict__ R) {
  const int co0  = blockIdx.x << 4;
  const int b    = blockIdx.y;          // 0..5  (= 2*branch + sub)
  const int tid  = threadIdx.x;
  const int wave = tid >> 5;
  const int lane = tid & 31;
  const int lh   = lane >> 4;
  const int lm   = lane & 15;

  __shared__ float red[8 * 256];

  v8f acc[3] = {};

  // lane's weight row, pre-offset by the half-wave K pairing
  const float* wrow =
      W + ((size_t)b * CH + (size_t)(co0 + lm)) * KTOT + 2 * lh;
  const float2* zf = (const float2*)Zcol;

  const int per_wave = NCH4 / 8;                      // 576 chunks
  const int k4beg    = wave * per_wave;
  for (int i = 0; i < per_wave; ++i) {
    const int k4 = k4beg + i;
    const int k  = k4 << 2;
    if ((i & 7) == 0)
      __builtin_prefetch(wrow + k + 1024, 0, 1);      // ~4 KB ahead in W stream
    const float2 aw = *(const float2*)(wrow + k);
    v2f a;
    a.x = aw.x;
    a.y = aw.y;
#pragma unroll
    for (int nt = 0; nt < 3; ++nt) {
      const float2 bw = zf[(size_t)(k4 * 3 + nt) * 32 + lane];
      v2f bf;
      bf.x = bw.x;
      bf.y = bw.y;
      acc[nt] = wmma_f32x4(a, bf, acc[nt]);
    }
  }

  // cross-wave reduction, one n-tile at a time (8 KB LDS)
#pragma unroll 1
  for (int nt = 0; nt < 3; ++nt) {
#pragma unroll
    for (int r = 0; r < 8; ++r) red[wave * 256 + r * 32 + lane] = acc[nt][r];
    __syncthreads();
    float s = 0.f;
#pragma unroll
    for (int w = 0; w < 8; ++w) s += red[w * 256 + tid];
    const int r   = tid >> 5;
    const int ln  = tid & 31;
    const int row = r + ((ln >> 4) << 3);   // C layout: M = r + 8*(lane>=16)
    const int col = (nt << 4) + (ln & 15);
    R[((size_t)b * CH + (size_t)(co0 + row)) * NSP + col] = s;
    __syncthreads();
  }
}

// ---------------------------------------------------------------------------
// Kernel 2: per-channel  relu(IN(conv0)) + relu(IN(conv1 @ L))  -> QKV
// grid = (CH, 3), block = 64.
// ---------------------------------------------------------------------------
__global__ __launch_bounds__(64) void norm_kernel(
    const float* __restrict__ R, const float* __restrict__ L,
    float* __restrict__ QKV) {
  const int ch  = blockIdx.x;
  const int b3  = blockIdx.y;            // 0=Q 1=K 2=V
  const int tid = threadIdx.x;
  __shared__ float s0[NSP], s1[NSP], stats[4];

  const float* r0 = R + ((size_t)(2 * b3)     * CH + ch) * NSP;
  const float* r1 = R + ((size_t)(2 * b3 + 1) * CH + ch) * NSP;

  if (tid < NSP) {
    s0[tid] = r0[tid];
    float a = 0.f;
    for (int k = 0; k < NSP; ++k) a += r1[k] * L[k * NSP + tid];  // row @ L
    s1[tid] = a;
  }
  __syncthreads();
  if (tid == 0) {
    float m = 0.f, v = 0.f;
    for (int k = 0; k < NSP; ++k) m += s0[k];
    m *= (1.0f / NSP);
    for (int k = 0; k < NSP; ++k) { float d = s0[k] - m; v += d * d; }
    stats[0] = m; stats[1] = v * (1.0f / NSP);
  }
  if (tid == 32) {
    float m = 0.f, v = 0.f;
    for (int k = 0; k < NSP; ++k) m += s1[k];
    m *= (1.0f / NSP);
    for (int k = 0; k < NSP; ++k) { float d = s1[k] - m; v += d * d; }
    stats[2] = m; stats[3] = v * (1.0f / NSP);
  }
  __syncthreads();
  if (tid < NSP) {
    float q0 = (s0[tid] - stats[0]) * rsqrtf(stats[1] + EPS);
    float q1 = (s1[tid] - stats[2]) * rsqrtf(stats[3] + EPS);
    q0 = fmaxf(q0, 0.f); q1 = fmaxf(q1, 0.f);
    QKV[((size_t)b3 * CH + ch) * NSP + tid] = q0 + q1;
  }
}

// ---------------------------------------------------------------------------
// Kernel 3: E = exp(Q @ K^T) via WMMA.  grid = (CH/16, CH/128), block = 256.
// The 128x48 K-tile is staged in LDS (async global->LDS when available);
// each wave then owns one 16x16 tile.
// ---------------------------------------------------------------------------
__global__ __launch_bounds__(256) void score_kernel(
    const float* __restrict__ Q, const float* __restrict__ Km,
    float* __restrict__ E) {
  const int tid  = threadIdx.x;
  const int wave = tid >> 5;
  const int lane = tid & 31;
  const int lh   = lane >> 4;
  const int lm   = lane & 15;
  const int i0   = blockIdx.x << 4;
  const int jb0  = blockIdx.y << 7;                  // 128-row K block

  __shared__ float sK[128 * NSP];                    // 24 KB

  {
    const float* gsrc = Km + (size_t)jb0 * NSP;      // 6144 floats, 16B aligned
#if defined(HAVE_ASYNC_LDS)
#pragma unroll
    for (int i = 0; i < 6; ++i) {
      const int idx = tid + i * 256;                 // float4 index, 0..1535
      __builtin_amdgcn_global_load_async_to_lds_b128(
          (ga_v4i*)(gsrc + idx * 4),
          (la_v4i*)&sK[idx * 4], 0, 0);
    }
    __builtin_amdgcn_s_wait_asynccnt(0);
    __syncthreads();
#else
#pragma unroll
    for (int i = 0; i < 6; ++i) {
      const int idx = tid + i * 256;
      ((float4*)sK)[idx] = ((const float4*)gsrc)[idx];
    }
    __syncthreads();
#endif
  }

  v8f acc = {};
  const float* qr = Q + (size_t)(i0 + lm) * NSP;     // A row for this lane
  const float* kl = sK + (size_t)(wave * 16 + lm) * NSP;  // B col (LDS)
#pragma unroll
  for (int k = 0; k < NSP; k += 4) {
    const float2 aw = *(const float2*)(qr + k + 2 * lh);
    const float2 bw = *(const float2*)(kl + k + 2 * lh);
    v2f a, bf;
    a.x = aw.x;  a.y = aw.y;
    bf.x = bw.x; bf.y = bw.y;
    acc = wmma_f32x4(a, bf, acc);
  }
#pragma unroll
  for (int r = 0; r < 8; ++r) {
    const int row = i0 + r + (lh << 3);
    const int col = jb0 + wave * 16 + lm;
    E[(size_t)row * CH + col] = __expf(acc[r]);
  }
}

// ---------------------------------------------------------------------------
// Kernel 4: row sums of E.  grid = CH, block = 256.
// ---------------------------------------------------------------------------
__global__ __launch_bounds__(256) void rowsum_kernel(
    const float* __restrict__ E, float* __restrict__ d) {
  __shared__ float sred[256];
  const int row = blockIdx.x;
  float s = 0.f;
  for (int j = threadIdx.x; j < CH; j += 256) s += E[(size_t)row * CH + j];
  sred[threadIdx.x] = s;
  __syncthreads();
  for (int off = 128; off > 0; off >>= 1) {
    if (threadIdx.x < off) sred[threadIdx.x] += sred[threadIdx.x + off];
    __syncthreads();
  }
  if (threadIdx.x == 0) d[row] = sred[0];
}

// ---------------------------------------------------------------------------
// Kernel 5: out[i,n] = sum_j 0.5*(E[i,j]/d[i] + E[j,i]/d[j]) * V[j,n]
// grid = (CH/16, 3), block = 256 (8 waves split the j reduction).
// ---------------------------------------------------------------------------
__global__ __launch_bounds__(256) void out_kernel(
    const float* __restrict__ E, const float* __restrict__ d,
    const float* __restrict__ V, float* __restrict__ out) {
  const int tid  = threadIdx.x;
  const int wave = tid >> 5;
  const int lane = tid & 31;
  const int lh   = lane >> 4;
  const int lm   = lane & 15;
  const int i0   = blockIdx.x << 4;
  const int n0   = blockIdx.y << 4;

  __shared__ float red[8 * 256];

  v8f acc = {};
  const int   m   = i0 + lm;            // A row owned by this lane
  const float rdm = 1.0f / d[m];

  for (int it = 0; it < CH / 32; ++it) {
    const int j  = (it << 5) + (wave << 2);
    const int ja = j + 2 * lh;
    v2f a, bf;
    {
      const float e1 = E[(size_t)m * CH + ja];
      const float e2 = E[(size_t)ja * CH + m];
      a.x = 0.5f * (e1 * rdm + e2 / d[ja]);
    }
    {
      const float e1 = E[(size_t)m * CH + (ja + 1)];
      const float e2 = E[(size_t)(ja + 1) * CH + m];
      a.y = 0.5f * (e1 * rdm + e2 / d[ja + 1]);
    }
    bf.x = V[(size_t)ja       * NSP + n0 + lm];
    bf.y = V[(size_t)(ja + 1) * NSP + n0 + lm];
    acc = wmma_f32x4(a, bf, acc);
  }

#pragma unroll
  for (int r = 0; r < 8; ++r) red[wave * 256 + r * 32 + lane] = acc[r];
  __syncthreads();
  float s = 0.f;
#pragma unroll
  for (int w = 0; w < 8; ++w) s += red[w * 256 + tid];
  const int r   = tid >> 5;
  const int ln  = tid & 31;
  const int row = i0 + r + ((ln >> 4) << 3);
  const int col = n0 + (ln & 15);
  out[(size_t)row * NSP + col] = s;
}

// ---------------------------------------------------------------------------
extern "C" void kernel_launch(void* const* d_in, const int* in_sizes, int n_in,
                              void* d_out, int out_size, void* d_ws,
                              size_t ws_size, hipStream_t stream) {
  const float* Z = (const float*)d_in[0];   // [2048, 48]
  const float* L = (const float*)d_in[1];   // [48, 48]
  const float* W = (const float*)d_in[2];   // [3, 2, 2048, 2048, 9]
  float* out = (float*)d_out;               // [2048, 48]

  float* ws   = (float*)d_ws;
  float* Zcol = ws;                                   // KTOT*NSP     (3.5 MB)
  float* R    = Zcol + (size_t)KTOT * NSP;            // 6*CH*NSP
  float* QKV  = R + (size_t)6 * CH * NSP;             // 3*CH*NSP
  float* E    = QKV + (size_t)3 * CH * NSP;           // CH*CH        (16 MB)
  float* dsum = E + (size_t)CH * CH;                  // CH

  im2col_kernel<<<(NCH4 * 3 * 32 + 255) / 256, 256, 0, stream>>>(Z, Zcol);
  conv_wmma_kernel<<<dim3(CH / 16, 6), 256, 0, stream>>>(W, Zcol, R);
  norm_kernel<<<dim3(CH, 3), 64, 0, stream>>>(R, L, QKV);
  score_kernel<<<dim3(CH / 16, CH / 128), 256, 0, stream>>>(
      QKV, QKV + (size_t)CH * NSP, E);
  rowsum_kernel<<<CH, 256, 0, stream>>>(E, dsum);
  out_kernel<<<dim3(CH / 16, 3), 256, 0, stream>>>(
      E, dsum, QKV + (size_t)2 * CH * NSP, out);
}